// AttLstm_42408507081044
// MI455X (gfx1250) — compile-verified
//
#include <hip/hip_runtime.h>
#include <hip/hip_bf16.h>
#include <math.h>

// ---------------------------------------------------------------------------
// Sizes (fixed by the reference)
// ---------------------------------------------------------------------------
#define S_LEN 512
#define HDIM  1024
#define IDIM  1024
#define G4    (4 * HDIM)   // 4096

typedef __attribute__((ext_vector_type(2))) float v2f;
typedef __attribute__((ext_vector_type(8))) float v8f;

__device__ __forceinline__ float sigmoidf_(float x) {
    return 1.0f / (1.0f + __expf(-x));
}

__device__ __forceinline__ float wave_reduce_add(float v) {
    // wave32 reduction
    #pragma unroll
    for (int off = 16; off > 0; off >>= 1)
        v += __shfl_down(v, off, 32);
    return v;
}

// ---------------------------------------------------------------------------
// WMMA f32 GEMM:  C[M x N] = A[M x K] * W[N x K]^T  (+ b1[n] + b2[n])
// One wave computes one 16x16 tile of C.  K must be a multiple of 4,
// M and N multiples of 16, grid sized exactly (EXEC all-ones for WMMA).
//
// A-matrix 16x4 f32 layout (ISA 7.12.2): lane L<16 holds {A[L][k], A[L][k+1]},
// lane L>=16 holds {A[L-16][k+2], A[L-16][k+3]}.  B (4x16) mirrors this with
// rows striped across lanes, so B frag lane L<16 = {W[n0+L][k], W[n0+L][k+1]},
// lane L>=16 = {W[n0+L-16][k+2], W[n0+L-16][k+3]}  (B = W^T).
// C/D: VGPR r -> row m0 + r + 8*(lane>>4), col n0 + (lane&15).
// ---------------------------------------------------------------------------
__global__ void wmma_gemm_nt_bias(const float* __restrict__ A,
                                  const float* __restrict__ W,
                                  const float* __restrict__ b1,
                                  const float* __restrict__ b2,
                                  float* __restrict__ C,
                                  int N, int K)
{
    const int wave = threadIdx.x >> 5;
    const int lane = threadIdx.x & 31;
    const int tiles_n = N >> 4;
    const int tile = blockIdx.x * 8 + wave;   // 8 waves / block
    const int tm = tile / tiles_n;
    const int tn = tile - tm * tiles_n;
    const int m0 = tm << 4;
    const int n0 = tn << 4;
    const int l16  = lane & 15;
    const int half = lane >> 4;

    const float* __restrict__ arow = A + (size_t)(m0 + l16) * K + 2 * half;
    const float* __restrict__ brow = W + (size_t)(n0 + l16) * K + 2 * half;

    v8f acc = {};
    for (int k = 0; k < K; k += 4) {
        v2f av = *(const v2f*)(arow + k);
        v2f bv = *(const v2f*)(brow + k);
        acc = __builtin_amdgcn_wmma_f32_16x16x4_f32(
            /*neg_a=*/false, av, /*neg_b=*/false, bv,
            /*c_mod=*/(short)0, acc, /*reuse_a=*/false, /*reuse_b=*/false);
    }

    float bias = 0.0f;
    if (b1) bias += b1[n0 + l16];
    if (b2) bias += b2[n0 + l16];

    float* cptr = C + (size_t)(m0 + 8 * half) * N + n0 + l16;
    #pragma unroll
    for (int r = 0; r < 8; ++r)
        cptr[(size_t)r * N] = acc[r] + bias;
}

// ---------------------------------------------------------------------------
// One LSTM cell step (batch = 1).  One wave per hidden index j in [0,1024).
// gates rows: i=j, f=j+1024, g=j+2048, o=j+3072.
//   g[row] = (gbase?gbase[row]:0) + (b1?b1[row]:0) + (b2?b2[row]:0)
//          + (xvec? x.Wih[row] : 0) + h.Whh[row]
// Writes c_out[j], h_out[j], out_row[j] = h2.
// Grid: 128 blocks x 256 threads = 1024 waves.
// ---------------------------------------------------------------------------
__global__ void lstm_step(const float* __restrict__ gbase,
                          const float* __restrict__ b1,
                          const float* __restrict__ b2,
                          const float* __restrict__ xvec,
                          const float* __restrict__ Wih,
                          const float* __restrict__ hvec,
                          const float* __restrict__ Whh,
                          const float* __restrict__ c_in,
                          float* __restrict__ c_out,
                          float* __restrict__ h_out,
                          float* __restrict__ out_row)
{
    const int j    = (blockIdx.x * blockDim.x + threadIdx.x) >> 5;
    const int lane = threadIdx.x & 31;

    const float* __restrict__ r0 = Whh + (size_t)j * HDIM;
    const float* __restrict__ r1 = r0 + (size_t)HDIM * HDIM;
    const float* __restrict__ r2 = r1 + (size_t)HDIM * HDIM;
    const float* __restrict__ r3 = r2 + (size_t)HDIM * HDIM;

    float a0 = 0.f, a1 = 0.f, a2 = 0.f, a3 = 0.f;

    if (xvec) {
        const float* __restrict__ q0 = Wih + (size_t)j * HDIM;
        const float* __restrict__ q1 = q0 + (size_t)HDIM * HDIM;
        const float* __restrict__ q2 = q1 + (size_t)HDIM * HDIM;
        const float* __restrict__ q3 = q2 + (size_t)HDIM * HDIM;
        for (int k = lane; k < HDIM; k += 32) {
            const float h = hvec[k];
            const float x = xvec[k];
            a0 += h * r0[k] + x * q0[k];
            a1 += h * r1[k] + x * q1[k];
            a2 += h * r2[k] + x * q2[k];
            a3 += h * r3[k] + x * q3[k];
        }
    } else {
        for (int k = lane; k < HDIM; k += 32) {
            const float h = hvec[k];
            a0 += h * r0[k];
            a1 += h * r1[k];
            a2 += h * r2[k];
            a3 += h * r3[k];
        }
    }

    a0 = wave_reduce_add(a0);
    a1 = wave_reduce_add(a1);
    a2 = wave_reduce_add(a2);
    a3 = wave_reduce_add(a3);

    if (lane == 0) {
        float gi = a0, gf = a1, gg = a2, go = a3;
        if (gbase) { gi += gbase[j];        gf += gbase[j + HDIM];
                     gg += gbase[j + 2*HDIM]; go += gbase[j + 3*HDIM]; }
        if (b1)    { gi += b1[j];           gf += b1[j + HDIM];
                     gg += b1[j + 2*HDIM];  go += b1[j + 3*HDIM]; }
        if (b2)    { gi += b2[j];           gf += b2[j + HDIM];
                     gg += b2[j + 2*HDIM];  go += b2[j + 3*HDIM]; }

        const float c  = c_in[j];
        const float c2 = sigmoidf_(gf) * c + sigmoidf_(gi) * tanhf(gg);
        const float h2 = sigmoidf_(go) * tanhf(c2);
        c_out[j]   = c2;
        h_out[j]   = h2;
        out_row[j] = h2;
    }
}

// ---------------------------------------------------------------------------
// Generic matvec: out[n] = bias[n] + sum_k W[n][k] * vin[k]
// One wave per output row.  Grid: (N/8) blocks x 256 threads.
// ---------------------------------------------------------------------------
__global__ void matvec_bias(const float* __restrict__ W,
                            const float* __restrict__ vin,
                            const float* __restrict__ bias,
                            float* __restrict__ out,
                            int K)
{
    const int n    = (blockIdx.x * blockDim.x + threadIdx.x) >> 5;
    const int lane = threadIdx.x & 31;
    const float* __restrict__ row = W + (size_t)n * K;
    float acc = 0.f;
    for (int k = lane; k < K; k += 32)
        acc += row[k] * vin[k];
    acc = wave_reduce_add(acc);
    if (lane == 0)
        out[n] = acc + (bias ? bias[n] : 0.0f);
}

// ---------------------------------------------------------------------------
// logits[i] = v2s_b + sum_j tanh(eh[i][j] + a[j]) * v[j]
// One wave per i (512 waves).  Grid: 64 blocks x 256 threads.
// ---------------------------------------------------------------------------
__global__ void att_logits(const float* __restrict__ eh,
                           const float* __restrict__ a,
                           const float* __restrict__ v,
                           const float* __restrict__ vb,
                           float* __restrict__ logits)
{
    const int i    = (blockIdx.x * blockDim.x + threadIdx.x) >> 5;
    const int lane = threadIdx.x & 31;
    const float* __restrict__ row = eh + (size_t)i * HDIM;
    float acc = 0.f;
    for (int j = lane; j < HDIM; j += 32)
        acc += tanhf(row[j] + a[j]) * v[j];
    acc = wave_reduce_add(acc);
    if (lane == 0)
        logits[i] = acc + vb[0];
}

// ---------------------------------------------------------------------------
// softmax over 512 logits, single block of 512 threads
// ---------------------------------------------------------------------------
__global__ void softmax512(const float* __restrict__ logits,
                           float* __restrict__ s)
{
    __shared__ float buf[S_LEN];
    const int t = threadIdx.x;
    const float x = logits[t];
    buf[t] = x;
    __syncthreads();
    for (int off = S_LEN / 2; off > 0; off >>= 1) {
        if (t < off) buf[t] = fmaxf(buf[t], buf[t + off]);
        __syncthreads();
    }
    const float m = buf[0];
    __syncthreads();
    const float e = __expf(x - m);
    buf[t] = e;
    __syncthreads();
    for (int off = S_LEN / 2; off > 0; off >>= 1) {
        if (t < off) buf[t] += buf[t + off];
        __syncthreads();
    }
    s[t] = e / buf[0];
}

// ---------------------------------------------------------------------------
// z[j] = sum_i s[i] * e_out[i][j]   (columns; coalesced across threads)
// Grid: 4 blocks x 256 threads.
// ---------------------------------------------------------------------------
__global__ void att_context(const float* __restrict__ s,
                            const float* __restrict__ e_out,
                            float* __restrict__ z)
{
    const int j = blockIdx.x * blockDim.x + threadIdx.x;  // 0..1023
    float acc = 0.f;
    for (int i = 0; i < S_LEN; ++i)
        acc += s[i] * e_out[(size_t)i * HDIM + j];
    z[j] = acc;
}

// ---------------------------------------------------------------------------
// Host-side orchestration
// ---------------------------------------------------------------------------
extern "C" void kernel_launch(void* const* d_in, const int* in_sizes, int n_in,
                              void* d_out, int out_size, void* d_ws, size_t ws_size,
                              hipStream_t stream)
{
    (void)in_sizes; (void)n_in; (void)out_size; (void)ws_size;

    const float* x        = (const float*)d_in[0];   // (512,1,1024)
    const float* e_Wih    = (const float*)d_in[1];   // (4096,1024)
    const float* e_Whh    = (const float*)d_in[2];   // (4096,1024)
    const float* e_bih    = (const float*)d_in[3];   // (4096)
    const float* e_bhh    = (const float*)d_in[4];   // (4096)
    const float* att_eh_W = (const float*)d_in[5];   // (1024,1024)
    const float* att_eh_b = (const float*)d_in[6];   // (1024)
    const float* att_ch_W = (const float*)d_in[7];   // (1024,1024)
    const float* att_ch_b = (const float*)d_in[8];   // (1024)
    const float* v2s_W    = (const float*)d_in[9];   // (1,1024)
    const float* v2s_b    = (const float*)d_in[10];  // (1)
    const float* att_bias = (const float*)d_in[11];  // (1024)
    const float* d_Wih    = (const float*)d_in[12];  // (4096,1024)
    const float* d_Whh    = (const float*)d_in[13];  // (4096,1024)
    const float* d_bih    = (const float*)d_in[14];  // (4096)
    const float* d_bhh    = (const float*)d_in[15];  // (4096)

    float* out = (float*)d_out;                      // (512,1,1024)

    // --- workspace carve-out (floats) ---
    float* ws = (float*)d_ws;
    float* Xg     = ws;                  ws += (size_t)S_LEN * G4;     // 512x4096
    float* e_out  = ws;                  ws += (size_t)S_LEN * HDIM;   // 512x1024
    float* eh     = ws;                  ws += (size_t)S_LEN * HDIM;   // 512x1024
    float* hbuf0  = ws;                  ws += HDIM;
    float* hbuf1  = ws;                  ws += HDIM;
    float* cbuf0  = ws;                  ws += HDIM;
    float* cbuf1  = ws;                  ws += HDIM;
    float* avec   = ws;                  ws += HDIM;
    float* zvec   = ws;                  ws += HDIM;
    float* logits = ws;                  ws += S_LEN;
    float* smax   = ws;                  ws += S_LEN;

    // ---------------- 1) Xg = x @ e_Wih^T + e_bih + e_bhh  (WMMA GEMM) ------
    {
        const int tiles = (S_LEN / 16) * (G4 / 16);   // 8192
        wmma_gemm_nt_bias<<<tiles / 8, 256, 0, stream>>>(
            x, e_Wih, e_bih, e_bhh, Xg, G4, IDIM);
    }

    // ---------------- 2) encoder recurrence (sequential) --------------------
    hipMemsetAsync(hbuf0, 0, HDIM * sizeof(float), stream);
    hipMemsetAsync(cbuf0, 0, HDIM * sizeof(float), stream);
    for (int t = 0; t < S_LEN; ++t) {
        float* h_in  = (t & 1) ? hbuf1 : hbuf0;
        float* h_out = (t & 1) ? hbuf0 : hbuf1;
        float* c_in  = (t & 1) ? cbuf1 : cbuf0;
        float* c_out = (t & 1) ? cbuf0 : cbuf1;
        lstm_step<<<128, 256, 0, stream>>>(
            Xg + (size_t)t * G4, nullptr, nullptr,
            nullptr, nullptr,
            h_in, e_Whh,
            c_in, c_out, h_out, e_out + (size_t)t * HDIM);
    }

    // ---------------- 3) eh = e_out @ att_eh_W^T + att_eh_b + att_bias ------
    {
        const int tiles = (S_LEN / 16) * (HDIM / 16);  // 2048
        wmma_gemm_nt_bias<<<tiles / 8, 256, 0, stream>>>(
            e_out, att_eh_W, att_eh_b, att_bias, eh, HDIM, HDIM);
    }

    // ---------------- 4) decoder recurrence (sequential) --------------------
    hipMemsetAsync(hbuf0, 0, HDIM * sizeof(float), stream);   // ctx0 = 0
    hipMemsetAsync(cbuf0, 0, HDIM * sizeof(float), stream);
    for (int t = 0; t < S_LEN; ++t) {
        float* ctx   = (t & 1) ? hbuf1 : hbuf0;
        float* h_out = (t & 1) ? hbuf0 : hbuf1;
        float* c_in  = (t & 1) ? cbuf1 : cbuf0;
        float* c_out = (t & 1) ? cbuf0 : cbuf1;

        // a = ctx @ att_ch_W^T + att_ch_b     (1024 waves)
        matvec_bias<<<HDIM / 8, 256, 0, stream>>>(att_ch_W, ctx, att_ch_b,
                                                  avec, HDIM);
        // logits[i] = sum_j tanh(eh[i][j]+a[j]) * v[j] + vb   (512 waves)
        att_logits<<<S_LEN / 8, 256, 0, stream>>>(eh, avec, v2s_W, v2s_b,
                                                  logits);
        // s = softmax(logits)
        softmax512<<<1, S_LEN, 0, stream>>>(logits, smax);
        // z = s @ e_out
        att_context<<<HDIM / 256, 256, 0, stream>>>(smax, e_out, zvec);
        // (h2, c2) = lstm_cell(z, ctx, c)
        lstm_step<<<128, 256, 0, stream>>>(
            nullptr, d_bih, d_bhh,
            zvec, d_Wih,
            ctx, d_Whh,
            c_in, c_out, h_out, out + (size_t)t * HDIM);
    }
}